// alphaC0_42C_4_19026705121515
// MI455X (gfx1250) — compile-verified
//
#include <hip/hip_runtime.h>

// Problem constants from the reference: B=2, G=64.
#define GRID_G 64
#define VOX    (GRID_G * GRID_G * GRID_G)   // 262144 = 2^18 voxels per batch
#define PAIRS_SHIFT 17                      // VOX/2 voxel-pairs per batch

typedef float v2f __attribute__((ext_vector_type(2)));

// 32-bit LDS byte address of a __shared__ object (generic -> AS3 -> int).
__device__ __forceinline__ unsigned lds_byte_addr(const void* p) {
    return (unsigned)(uintptr_t)(__attribute__((address_space(3))) const void*)p;
}

__device__ __forceinline__ v2f splat2(float x) { return v2f{x, x}; }

// Two voxels (one float2) per thread. A = per-voxel 3x3 matrix, C0 fixed 3^4.
// C[b,i,j,k,l,v] = sum_{mnop} A[m,i]A[n,j]A[o,k]A[p,l] C0[m,n,o,p]
__global__ __launch_bounds__(256) void alphaC0_c4_kernel(
    const float* __restrict__ a,      // (B,3,3,G,G,G)
    const float* __restrict__ c0,     // (3,3,3,3)
    float* __restrict__ out)          // (B,3,3,3,3,G,G,G)
{
    __shared__ v2f lds_a[9 * 256];    // 18 KB

    const int      tid  = threadIdx.x;
    const unsigned gtid = blockIdx.x * 256u + (unsigned)tid;
    const unsigned b    = gtid >> PAIRS_SHIFT;            // uniform per block
    const unsigned vp   = gtid & ((1u << PAIRS_SHIFT) - 1u);

    // Base address of this thread's voxel pair within batch b.
    const float* abase = a + (size_t)b * 9u * (size_t)VOX + (size_t)vp * 2u;

    // ---- CDNA5 async path (GV form, proven syntax): 9 async global->LDS
    // b64 copies (one per (m,i) plane), overlapping the scalar C0 loads.
#pragma unroll
    for (int mi = 0; mi < 9; ++mi) {
        unsigned laddr = lds_byte_addr(&lds_a[mi * 256 + tid]);
        unsigned long long gaddr =
            (unsigned long long)(uintptr_t)(abase + (size_t)mi * VOX);
        asm volatile("global_load_async_to_lds_b64 %0, %1, off"
                     :: "v"(laddr), "v"(gaddr)
                     : "memory");
    }

    // ---- C0 is grid-uniform: constant-indexed loads through __restrict
    // const pointer -> s_load into SGPRs (free scalar FMA operands).
    float C0r[81];
#pragma unroll
    for (int q = 0; q < 81; ++q) C0r[q] = c0[q];

    // Wait for this wave's async LDS writes, then read back our 9 A pairs.
    asm volatile("s_wait_asynccnt 0x0" ::: "memory");

    v2f A[9];   // A[m*3 + i], two voxels wide
#pragma unroll
    for (int mi = 0; mi < 9; ++mi) A[mi] = lds_a[mi * 256 + tid];

    const size_t outbase = (size_t)b * 81u * (size_t)VOX + (size_t)vp * 2u;

    // Staged contraction: 4 stages x 243 MACs x 2 voxels, fully unrolled.
#pragma unroll
    for (int i = 0; i < 3; ++i) {
        v2f t1[27];                        // t1[n*9 + o*3 + p]
#pragma unroll
        for (int nop = 0; nop < 27; ++nop) {
            v2f s = A[0 + i] * splat2(C0r[0 + nop]);                     // m=0
            s = __builtin_elementwise_fma(A[3 + i], splat2(C0r[27 + nop]), s);
            s = __builtin_elementwise_fma(A[6 + i], splat2(C0r[54 + nop]), s);
            t1[nop] = s;
        }
#pragma unroll
        for (int j = 0; j < 3; ++j) {
            v2f t2[9];                     // t2[o*3 + p]
#pragma unroll
            for (int op = 0; op < 9; ++op) {
                v2f s = A[0 + j] * t1[0 + op];                           // n=0
                s = __builtin_elementwise_fma(A[3 + j], t1[9 + op], s);
                s = __builtin_elementwise_fma(A[6 + j], t1[18 + op], s);
                t2[op] = s;
            }
#pragma unroll
            for (int k = 0; k < 3; ++k) {
                v2f t3[3];                 // t3[p]
#pragma unroll
                for (int p = 0; p < 3; ++p) {
                    v2f s = A[0 + k] * t2[0 + p];                        // o=0
                    s = __builtin_elementwise_fma(A[3 + k], t2[3 + p], s);
                    s = __builtin_elementwise_fma(A[6 + k], t2[6 + p], s);
                    t3[p] = s;
                }
#pragma unroll
                for (int l = 0; l < 3; ++l) {
                    v2f s = A[0 + l] * t3[0];                            // p=0
                    s = __builtin_elementwise_fma(A[3 + l], t3[1], s);
                    s = __builtin_elementwise_fma(A[6 + l], t3[2], s);
                    const unsigned idx = (unsigned)(((i * 3 + j) * 3 + k) * 3 + l);
                    // Write-once 170 MB stream: non-temporal b64 store.
                    __builtin_nontemporal_store(
                        s, (v2f*)(out + outbase + (size_t)idx * VOX));
                }
            }
        }
    }
}

extern "C" void kernel_launch(void* const* d_in, const int* in_sizes, int n_in,
                              void* d_out, int out_size, void* d_ws, size_t ws_size,
                              hipStream_t stream) {
    const float* a  = (const float*)d_in[0];   // (B,3,3,G,G,G) fp32
    const float* c0 = (const float*)d_in[1];   // (3,3,3,3) fp32
    float* out = (float*)d_out;

    const int B = in_sizes[0] / (9 * VOX);     // = 2 for the reference
    const unsigned total_threads = (unsigned)B * (unsigned)(VOX / 2); // 2 voxels/thread
    dim3 grid(total_threads / 256u), block(256u);

    hipLaunchKernelGGL(alphaC0_c4_kernel, grid, block, 0, stream, a, c0, out);
}